// BaseModel_88055419503307
// MI455X (gfx1250) — compile-verified
//
#include <hip/hip_runtime.h>

typedef __attribute__((ext_vector_type(2))) float v2f;
typedef __attribute__((ext_vector_type(8))) float v8f;

// ---------------------------------------------------------------------------
// copy kernel: agg = src  (self term of GIN aggregation), float4-vectorized
// ---------------------------------------------------------------------------
__global__ __launch_bounds__(256) void copy_f4_kernel(const float4* __restrict__ src,
                                                      float4* __restrict__ dst,
                                                      long long n4) {
    long long t = (long long)blockIdx.x * blockDim.x + threadIdx.x;
    if (t < n4) dst[t] = src[t];
}

// ---------------------------------------------------------------------------
// edge scatter-add: agg[dst] += feat[src], CH channels (CH = 128 or 64)
// thread = (edge, 4-channel group); gathers coalesce per edge row
// ---------------------------------------------------------------------------
template <int CH>
__global__ __launch_bounds__(256) void scatter_add_kernel(const float* __restrict__ feat,
                                                          float* __restrict__ agg,
                                                          const int* __restrict__ srcIdx,
                                                          const int* __restrict__ dstIdx,
                                                          int n_edges) {
    constexpr int GROUPS = CH / 4;            // power of two (32 or 16)
    constexpr int SHIFT  = (GROUPS == 32) ? 5 : 4;
    long long t = (long long)blockIdx.x * blockDim.x + threadIdx.x;
    long long total = (long long)n_edges * GROUPS;
    if (t >= total) return;
    int e  = (int)(t >> SHIFT);
    int c4 = (int)(t & (GROUPS - 1));
    int s = srcIdx[e];
    int d = dstIdx[e];
    float4 v = reinterpret_cast<const float4*>(feat + (size_t)s * CH)[c4];
    float* dp = agg + (size_t)d * CH + (size_t)c4 * 4;
    atomicAdd(dp + 0, v.x);
    atomicAdd(dp + 1, v.y);
    atomicAdd(dp + 2, v.z);
    atomicAdd(dp + 3, v.w);
}

// ---------------------------------------------------------------------------
// WMMA f32 GEMM: C[n x NC] = relu(A[n x K] @ W[K x NC] + bias)
// One wave -> one 16-row tile, all NC columns. W, bias staged in LDS.
// Uses V_WMMA_F32_16X16X4_F32 (full f32 matrix pipe, K-step 4).
// A-fragment layout (16x4 f32): lane = 16*half + m, VGPR0 = (M=m, K=kb+2*half),
// VGPR1 = (M=m, K=kb+2*half+1).  C/D: VGPR v -> (M = v + 8*half, N = m).
// ---------------------------------------------------------------------------
template <int K, int NC>
__global__ __launch_bounds__(256) void gemm_bias_relu_kernel(const float* __restrict__ A,
                                                             const float* __restrict__ W,
                                                             const float* __restrict__ bias,
                                                             float* __restrict__ C,
                                                             int n) {
    __shared__ float sW[K * NC];
    __shared__ float sB[NC];
    for (int i = threadIdx.x; i < K * NC; i += blockDim.x) sW[i] = W[i];
    if (threadIdx.x < NC) sB[threadIdx.x] = bias[threadIdx.x];
    __syncthreads();

    const int wave = threadIdx.x >> 5;
    const int lane = threadIdx.x & 31;
    const int half = lane >> 4;   // 0: lanes 0-15, 1: lanes 16-31
    const int m    = lane & 15;

    const int row_base = (blockIdx.x * 8 + wave) * 16;
    if (row_base >= n) return;    // wave-uniform: EXEC stays all-ones for WMMA

    // Cache all A fragments for this 16-row tile in VGPRs (reused per N-tile)
    v2f afrag[K / 4];
    const float* arow = A + (size_t)(row_base + m) * K + 2 * half;
#pragma unroll
    for (int kb = 0; kb < K; kb += 4) {
        afrag[kb / 4].x = arow[kb + 0];
        afrag[kb / 4].y = arow[kb + 1];
    }

#pragma unroll
    for (int nt = 0; nt < NC / 16; ++nt) {
        v8f acc = {0.f, 0.f, 0.f, 0.f, 0.f, 0.f, 0.f, 0.f};
#pragma unroll
        for (int kb = 0; kb < K; kb += 4) {
            const int k0 = kb + 2 * half;
            v2f b;
            b.x = sW[(k0 + 0) * NC + nt * 16 + m];
            b.y = sW[(k0 + 1) * NC + nt * 16 + m];
            acc = __builtin_amdgcn_wmma_f32_16x16x4_f32(
                false, afrag[kb / 4], false, b, (short)0, acc, false, false);
        }
        const float bcol = sB[nt * 16 + m];
#pragma unroll
        for (int v = 0; v < 8; ++v) {
            int row = row_base + v + 8 * half;
            float val = acc[v] + bcol;
            val = val > 0.f ? val : 0.f;
            C[(size_t)row * NC + nt * 16 + m] = val;
        }
    }
}

// ---------------------------------------------------------------------------
// Fused head: h3 = relu(H[n x 64] @ W3[64 x 16] + b3); out = h3 @ W4 + b4
// WMMA for the 64->16 GEMM, then cross-lane shuffle reduction for 16->1.
// ---------------------------------------------------------------------------
__global__ __launch_bounds__(256) void head_kernel(const float* __restrict__ H,
                                                   const float* __restrict__ W3,
                                                   const float* __restrict__ b3,
                                                   const float* __restrict__ W4,
                                                   const float* __restrict__ b4,
                                                   float* __restrict__ out,
                                                   int n) {
    __shared__ float sW[64 * 16];
    __shared__ float sB[16];
    __shared__ float sW4[16];
    for (int i = threadIdx.x; i < 64 * 16; i += blockDim.x) sW[i] = W3[i];
    if (threadIdx.x < 16) {
        sB[threadIdx.x]  = b3[threadIdx.x];
        sW4[threadIdx.x] = W4[threadIdx.x];
    }
    __syncthreads();

    const int wave = threadIdx.x >> 5;
    const int lane = threadIdx.x & 31;
    const int half = lane >> 4;
    const int m    = lane & 15;

    const int row_base = (blockIdx.x * 8 + wave) * 16;
    if (row_base >= n) return;

    const float* arow = H + (size_t)(row_base + m) * 64 + 2 * half;
    v8f acc = {0.f, 0.f, 0.f, 0.f, 0.f, 0.f, 0.f, 0.f};
#pragma unroll
    for (int kb = 0; kb < 64; kb += 4) {
        v2f a;
        a.x = arow[kb + 0];
        a.y = arow[kb + 1];
        const int k0 = kb + 2 * half;
        v2f b;
        b.x = sW[(k0 + 0) * 16 + m];
        b.y = sW[(k0 + 1) * 16 + m];
        acc = __builtin_amdgcn_wmma_f32_16x16x4_f32(
            false, a, false, b, (short)0, acc, false, false);
    }

    const float bcol = sB[m];
    const float w4m  = sW4[m];
    const float b4v  = b4[0];
#pragma unroll
    for (int v = 0; v < 8; ++v) {
        float val = acc[v] + bcol;
        val = val > 0.f ? val : 0.f;
        float prod = val * w4m;
        // reduce over the 16 lanes of this half-wave (N dimension)
        prod += __shfl_xor(prod, 1);
        prod += __shfl_xor(prod, 2);
        prod += __shfl_xor(prod, 4);
        prod += __shfl_xor(prod, 8);
        if (m == 0) out[row_base + v + 8 * half] = prod + b4v;
    }
}

// ---------------------------------------------------------------------------
// Launcher
// ---------------------------------------------------------------------------
extern "C" void kernel_launch(void* const* d_in, const int* in_sizes, int n_in,
                              void* d_out, int out_size, void* d_ws, size_t ws_size,
                              hipStream_t stream) {
    const float* x  = (const float*)d_in[0];
    const int*   ei = (const int*)d_in[1];
    const float* W1 = (const float*)d_in[2];
    const float* b1 = (const float*)d_in[3];
    const float* W2 = (const float*)d_in[4];
    const float* b2 = (const float*)d_in[5];
    const float* W3 = (const float*)d_in[6];
    const float* b3 = (const float*)d_in[7];
    const float* W4 = (const float*)d_in[8];
    const float* b4 = (const float*)d_in[9];

    const int n       = in_sizes[0] / 128;     // 100000 (multiple of 16)
    const int n_edges = in_sizes[1] / 2;       // 1600000
    const int* srcIdx = ei;                    // edge_index[0, :]
    const int* dstIdx = ei + n_edges;          // edge_index[1, :]

    float* ws   = (float*)d_ws;
    float* agg1 = ws;                          // n*128 floats (region A)
    float* h1   = ws + (size_t)n * 128;        // n*64  floats (region B)
    float* agg2 = agg1;                        // reuse A (agg1 dead after GEMM1)
    float* h2   = h1;                          // reuse B (h1 dead after scatter2)

    float* outp = (float*)d_out;

    const int TPB = 256;
    auto cdiv = [](long long a, long long b) { return (int)((a + b - 1) / b); };

    // ---- layer 1: agg1 = x + scatter_sum(x[src] -> dst) ----
    copy_f4_kernel<<<cdiv((long long)n * 32, TPB), TPB, 0, stream>>>(
        (const float4*)x, (float4*)agg1, (long long)n * 32);
    scatter_add_kernel<128><<<cdiv((long long)n_edges * 32, TPB), TPB, 0, stream>>>(
        x, agg1, srcIdx, dstIdx, n_edges);
    // h1 = relu(agg1 @ W1 + b1)
    gemm_bias_relu_kernel<128, 64><<<cdiv(n, 128), TPB, 0, stream>>>(
        agg1, W1, b1, h1, n);

    // ---- layer 2: agg2 = h1 + scatter_sum(h1[src] -> dst) ----
    copy_f4_kernel<<<cdiv((long long)n * 16, TPB), TPB, 0, stream>>>(
        (const float4*)h1, (float4*)agg2, (long long)n * 16);
    scatter_add_kernel<64><<<cdiv((long long)n_edges * 16, TPB), TPB, 0, stream>>>(
        h1, agg2, srcIdx, dstIdx, n_edges);
    // h2 = relu(agg2 @ W2 + b2)
    gemm_bias_relu_kernel<64, 64><<<cdiv(n, 128), TPB, 0, stream>>>(
        agg2, W2, b2, h2, n);

    // ---- head: out = relu(h2 @ W3 + b3) @ W4 + b4 ----
    head_kernel<<<cdiv(n, 128), TPB, 0, stream>>>(h2, W3, b3, W4, b4, outp, n);
}